// CenterPixelCrossAttention_4260607557881
// MI455X (gfx1250) — compile-verified
//
#include <hip/hip_runtime.h>
#include <hip/hip_bf16.h>
#include <math.h>

// Problem constants (match reference)
constexpr int BATCH = 16;
constexpr int NTOK  = 4096;   // N_V * N_H
constexpr int DIMC  = 512;
constexpr int HEADS = 8;
constexpr int DHEAD = 64;
constexpr int INNER = 512;

typedef float v2f __attribute__((ext_vector_type(2)));
typedef float v4f __attribute__((ext_vector_type(4)));
typedef float v8f __attribute__((ext_vector_type(8)));

// ---------------------------------------------------------------------------
// Kernel 1: q = sel @ Wq^T (per head), then r[b,h,c] = sum_d q[d]*Wk[h*64+d,c]
// sm_scale = 1/sqrt(DHEAD) folded into r. One block per (b,h), 64 threads.
// ---------------------------------------------------------------------------
__global__ void qr_kernel(const float* __restrict__ x,
                          const float* __restrict__ Wq,
                          const float* __restrict__ Wk,
                          const int*   __restrict__ pidx,
                          const int*   __restrict__ scale,
                          float* __restrict__ r) {
  const int b = blockIdx.x, h = blockIdx.y, t = threadIdx.x; // t in [0,64)
  __shared__ float sel[DIMC];
  __shared__ float q[DHEAD];

  const int idx = pidx[2 * b] * scale[1] + pidx[2 * b + 1];
  const float* xrow = x + ((size_t)b * NTOK + idx) * DIMC;
  for (int c = t; c < DIMC; c += 64) sel[c] = xrow[c];
  __syncthreads();

  // q[t] = sel . Wq[h*64+t, :]
  const float* wqr = Wq + (size_t)(h * DHEAD + t) * DIMC;
  float acc = 0.f;
  for (int c = 0; c < DIMC; ++c) acc = fmaf(sel[c], wqr[c], acc);
  q[t] = acc;
  __syncthreads();

  const float sm = 0.125f; // DHEAD^-0.5
  for (int c = t; c < DIMC; c += 64) {
    float rc = 0.f;
    for (int d = 0; d < DHEAD; ++d)
      rc = fmaf(q[d], Wk[(size_t)(h * DHEAD + d) * DIMC + c], rc);
    r[((size_t)b * HEADS + h) * DIMC + c] = rc * sm;
  }
}

// ---------------------------------------------------------------------------
// Kernel 2: dots[b,h,k] = r[b,h,:] . x[b,k,:]   via V_WMMA_F32_16X16X4_F32
// A (16x4) = r rows (heads, padded 8->16 by duplication; dup rows never stored)
// B (4x16) = 16 consecutive k-rows of x
// C (16x16) f32 accumulator over K = DIM in steps of 4 (128 WMMAs / wave)
// Block = 256 threads = 8 waves; each wave owns a 16-row k-tile.
// ---------------------------------------------------------------------------
__global__ void dots_kernel(const float* __restrict__ x,
                            const float* __restrict__ r,
                            float* __restrict__ dots) {
  const int b    = blockIdx.y;
  const int wave = threadIdx.x >> 5;
  const int lane = threadIdx.x & 31;
  const int k0   = blockIdx.x * 128 + wave * 16;
  const int m    = lane & 15;   // A: head row (mod 8); B: k-row within tile
  const int half = lane >> 4;   // selects K pair {0,1} vs {2,3}

  const float* rb = r + (size_t)(b * HEADS + (m & 7)) * DIMC; // A fragment row
  const float* xb = x + ((size_t)b * NTOK + k0 + m) * DIMC;   // B fragment row

  v8f c = {};
  for (int cc = 0; cc < DIMC; cc += 4) {
    v2f a  = *(const v2f*)(rb + cc + 2 * half);
    v2f bf = *(const v2f*)(xb + cc + 2 * half);
    c = __builtin_amdgcn_wmma_f32_16x16x4_f32(
        /*neg_a=*/false, a, /*neg_b=*/false, bf,
        /*c_mod=*/(short)0, c, /*reuse_a=*/false, /*reuse_b=*/false);
  }

  // C layout: VGPR v, lane l: M = v + 8*(l>>4), N = l&15.
  // Heads 0..7 live entirely in lanes 0..15 (M = v).
  if (lane < 16) {
    float* dp = dots + (size_t)b * HEADS * NTOK;
#pragma unroll
    for (int v = 0; v < 8; ++v)
      dp[(size_t)v * NTOK + k0 + m] = c[v];
  }
}

// ---------------------------------------------------------------------------
// Kernel 3: in-place softmax over k (N=4096) for each (b,h) row.
// ---------------------------------------------------------------------------
__global__ void softmax_kernel(float* __restrict__ dots) {
  float* row = dots + (size_t)blockIdx.x * NTOK;
  __shared__ float red[256];
  const int t = threadIdx.x;

  float mx = -INFINITY;
  for (int k = t; k < NTOK; k += 256) mx = fmaxf(mx, row[k]);
  red[t] = mx; __syncthreads();
  for (int s = 128; s > 0; s >>= 1) {
    if (t < s) red[t] = fmaxf(red[t], red[t + s]);
    __syncthreads();
  }
  mx = red[0]; __syncthreads();

  float sum = 0.f;
  for (int k = t; k < NTOK; k += 256) {
    float e = expf(row[k] - mx);
    row[k] = e;
    sum += e;
  }
  red[t] = sum; __syncthreads();
  for (int s = 128; s > 0; s >>= 1) {
    if (t < s) red[t] += red[t + s];
    __syncthreads();
  }
  const float inv = 1.f / red[0];
  for (int k = t; k < NTOK; k += 256) row[k] *= inv;
}

// ---------------------------------------------------------------------------
// Kernel 4: s[b,h,c] = sum_k attn[b,h,k] * x[b,k,c]
// Grid (DIM/64, B), 256 threads = 16 k-groups x 16 lanes; lane owns 4 c's.
// Each group streams 256 k-rows (b128 coalesced); attn staged in LDS
// transposed to [g][k][h] so the 8 head weights are 2 b128 LDS broadcasts.
// Fixed-order cross-group LDS reduction keeps the result deterministic.
// 128 blocks * 8 waves = 1024 waves; x read exactly once (L2-resident).
// ---------------------------------------------------------------------------
__global__ void sacc_kernel(const float* __restrict__ x,
                            const float* __restrict__ attn,
                            float* __restrict__ s) {
  const int b    = blockIdx.y;
  const int c0   = blockIdx.x * 64;
  const int t    = threadIdx.x;
  const int g    = t >> 4;   // k-group 0..15 (rows g*256 .. g*256+255)
  const int lane = t & 15;   // c sub-tile: columns c0 + lane*4 .. +3

  // 32 KB: phase 1 = attn chunk att[g][k' (64)][h (8)];
  // phase 2 = partials red[g][h*64 + c].
  __shared__ float lds[16 * 64 * 8];

  float acc[8][4] = {};
  const float* ab = attn + (size_t)b * HEADS * NTOK;
  const float* xb = x + (size_t)b * NTOK * DIMC + c0 + lane * 4;

  for (int kc = 0; kc < 256; kc += 64) {   // 4 chunks of 64 k' per group
    __syncthreads();
    // Cooperative fill: lds[g2*512 + k2*8 + h2] = attn[b, h2, g2*256 + kc + k2]
    // consecutive t -> consecutive k2 => coalesced 256B runs per (g2,h2).
#pragma unroll
    for (int j = 0; j < 8192; j += 256) {
      const int i  = j + t;
      const int k2 = i & 63;
      const int g2 = (i >> 6) & 15;
      const int h2 = i >> 10;
      lds[g2 * 512 + k2 * 8 + h2] = ab[(size_t)h2 * NTOK + g2 * 256 + kc + k2];
    }
    __syncthreads();

    const float* xg = xb + (size_t)(g * 256 + kc) * DIMC;
    const float* ag = lds + g * 512;
    for (int k2 = 0; k2 < 64; ++k2) {
      const v4f xv  = *(const v4f*)(xg + (size_t)k2 * DIMC);
      const v4f a03 = *(const v4f*)(ag + k2 * 8);
      const v4f a47 = *(const v4f*)(ag + k2 * 8 + 4);
#pragma unroll
      for (int h = 0; h < 4; ++h)
#pragma unroll
        for (int jj = 0; jj < 4; ++jj) {
          acc[h][jj]     = fmaf(a03[h], xv[jj], acc[h][jj]);
          acc[h + 4][jj] = fmaf(a47[h], xv[jj], acc[h + 4][jj]);
        }
    }
  }

  // Cross-group reduction (fixed order => deterministic).
  __syncthreads();
#pragma unroll
  for (int h = 0; h < 8; ++h)
#pragma unroll
    for (int jj = 0; jj < 4; ++jj)
      lds[g * 512 + h * 64 + lane * 4 + jj] = acc[h][jj];
  __syncthreads();

  for (int o = t; o < 512; o += 256) {      // o = h*64 + c
    const int h = o >> 6, c = o & 63;
    float sum = 0.f;
    for (int gg = 0; gg < 16; ++gg) sum += lds[gg * 512 + h * 64 + c];
    s[((size_t)b * HEADS + h) * DIMC + c0 + c] = sum;
  }
}

// ---------------------------------------------------------------------------
// Kernel 5: o_inner[i] = Wv[i,:] . s[b, i/64, :]; out[b,j] = bo[j] + Wo[j,:].o_inner
// One block per batch, 256 threads.
// ---------------------------------------------------------------------------
__global__ void out_kernel(const float* __restrict__ s,
                           const float* __restrict__ Wv,
                           const float* __restrict__ Wo,
                           const float* __restrict__ bo,
                           float* __restrict__ out) {
  const int b = blockIdx.x, t = threadIdx.x;
  __shared__ float sl[HEADS * DIMC]; // 16 KB
  __shared__ float oi[INNER];

  for (int i = t; i < HEADS * DIMC; i += 256)
    sl[i] = s[(size_t)b * HEADS * DIMC + i];
  __syncthreads();

  for (int i = t; i < INNER; i += 256) {
    const float* wv = Wv + (size_t)i * DIMC;
    const float* sh = sl + (i / DHEAD) * DIMC;
    float acc = 0.f;
    for (int cc = 0; cc < DIMC; ++cc) acc = fmaf(wv[cc], sh[cc], acc);
    oi[i] = acc;
  }
  __syncthreads();

  for (int j = t; j < DIMC; j += 256) {
    const float* wo = Wo + (size_t)j * INNER;
    float acc = bo[j];
    for (int i = 0; i < INNER; ++i) acc = fmaf(wo[i], oi[i], acc);
    out[(size_t)b * DIMC + j] = acc;
  }
}

// ---------------------------------------------------------------------------
extern "C" void kernel_launch(void* const* d_in, const int* in_sizes, int n_in,
                              void* d_out, int out_size, void* d_ws, size_t ws_size,
                              hipStream_t stream) {
  const float* x     = (const float*)d_in[0];
  // d_in[1] ("data") is unused by the reference.
  const float* Wq    = (const float*)d_in[2];
  const float* Wk    = (const float*)d_in[3];
  const float* Wv    = (const float*)d_in[4];
  const float* Wo    = (const float*)d_in[5];
  const float* bo    = (const float*)d_in[6];
  const int*   pidx  = (const int*)d_in[7];
  const int*   scale = (const int*)d_in[8];
  float* out = (float*)d_out;

  // Scratch layout (floats): r[B*H*DIM] | dots[B*H*N] | s[B*H*DIM]  (~2.6 MB)
  float* r    = (float*)d_ws;
  float* dots = r + (size_t)BATCH * HEADS * DIMC;
  float* s    = dots + (size_t)BATCH * HEADS * NTOK;

  qr_kernel<<<dim3(BATCH, HEADS), 64, 0, stream>>>(x, Wq, Wk, pidx, scale, r);
  dots_kernel<<<dim3(NTOK / 128, BATCH), 256, 0, stream>>>(x, r, dots);
  softmax_kernel<<<BATCH * HEADS, 256, 0, stream>>>(dots);
  sacc_kernel<<<dim3(DIMC / 64, BATCH), 256, 0, stream>>>(x, dots, s);
  out_kernel<<<BATCH, 256, 0, stream>>>(s, Wv, Wo, bo, out);
}